// GAT_83975200572033
// MI455X (gfx1250) — compile-verified
//
#include <hip/hip_runtime.h>
#include <hip/hip_bf16.h>

typedef __attribute__((ext_vector_type(2))) float v2f;
typedef __attribute__((ext_vector_type(8))) float v8f;

#define B_ 8
#define N_ 512
#define F_ 128
#define T_ 32
#define D_ 64
#define EPS_ 1e-6f

// ---------------------------------------------------------------------------
// Kernel 1: z[b,t,n,d] = tanh( sum_f x[b,n,f,t] * W[d,f] + bias[d] )
// One wave32 computes one 16(n) x 16(d) tile via V_WMMA_F32_16X16X4_F32.
// wave id bits: [1:0]=dtile(4) [6:2]=ntile(32) [11:7]=t(32) [14:12]=b(8)
// ---------------------------------------------------------------------------
__global__ void GAT_z_kernel(const float* __restrict__ x,
                             const float* __restrict__ W,
                             const float* __restrict__ bias,
                             float* __restrict__ z) {
  const int wave = blockIdx.x * (blockDim.x >> 5) + (threadIdx.x >> 5);
  const int lane = threadIdx.x & 31;

  const int dtile = wave & 3;
  const int ntile = (wave >> 2) & 31;
  const int t     = (wave >> 7) & 31;
  const int b     = wave >> 12;

  const int l15  = lane & 15;
  const int koff = (lane >> 4) << 1;     // K pair: lanes 0-15 -> {k,k+1}, 16-31 -> {k+2,k+3}
  const int mrow = ntile * 16 + l15;     // A-matrix row (n index)
  const int dcol = dtile * 16 + l15;     // B-matrix col (d index)

  // x[b][n][f][t] : element f lives at xrow[f * T_]
  const float* xrow = x + ((size_t)(b * N_ + mrow) * F_) * T_ + t;
  const float* wrow = W + (size_t)dcol * F_;   // W[d][f], contiguous over f

  v8f c = {};
  for (int k = 0; k < F_; k += 4) {
    v2f a, bm;
    a.x  = xrow[(size_t)(k + koff)     * T_];
    a.y  = xrow[(size_t)(k + koff + 1) * T_];
    bm.x = wrow[k + koff];
    bm.y = wrow[k + koff + 1];
    c = __builtin_amdgcn_wmma_f32_16x16x4_f32(false, a, false, bm,
                                              (short)0, c, false, false);
  }

  const float bv = bias[dcol];
  const int rbase = (lane >> 4) << 3;          // C/D: rows v (lanes<16) or v+8
  const size_t zbt = (size_t)(b * T_ + t) * N_;
#pragma unroll
  for (int v = 0; v < 8; ++v) {
    const int n = ntile * 16 + rbase + v;
    z[(zbt + n) * D_ + dcol] = tanhf(c[v] + bv);
  }
}

// ---------------------------------------------------------------------------
// Kernel 2: sq[b,t,n] = sum_d z^2
// ---------------------------------------------------------------------------
__global__ void GAT_sq_kernel(const float* __restrict__ z,
                              float* __restrict__ sq) {
  const int i = blockIdx.x * blockDim.x + threadIdx.x;
  if (i >= B_ * T_ * N_) return;
  const float* zr = z + (size_t)i * D_;
  float s = 0.f;
#pragma unroll 8
  for (int d = 0; d < D_; ++d) { const float v = zr[d]; s = fmaf(v, v, s); }
  sq[i] = s;
}

// ---------------------------------------------------------------------------
// Kernel 3: attention + softmax + mean over B.
// Block = (t, ntile): 16 rows x 512 cols, 256 threads (8 waves).
// Per b: dot via fp32 WMMA, fused cos/dist/exp -> LDS, row softmax, register
// accumulation of the 1/B mean. Single final store, no atomics.
// ---------------------------------------------------------------------------
__global__ void __launch_bounds__(256)
GAT_att_kernel(const float* __restrict__ z,
               const float* __restrict__ sq,
               float* __restrict__ out) {
  __shared__ float att_s[16 * 512];   // 32 KB
  __shared__ float sq_s[512];         //  2 KB
  __shared__ float red_s[256];        //  1 KB
  __shared__ float rowsum_s[16];

  const int tid   = threadIdx.x;
  const int wave  = tid >> 5;
  const int lane  = tid & 31;
  const int ntile = blockIdx.x & 31;
  const int t     = blockIdx.x >> 5;

  const int l15   = lane & 15;
  const int koff  = (lane >> 4) << 1;
  const int rbase = (lane >> 4) << 3;
  const int arow  = ntile * 16 + l15;      // A row (n), same for all 4 col tiles

  float acc[32];                           // 16*512 / 256 threads = 32 each
#pragma unroll
  for (int j = 0; j < 32; ++j) acc[j] = 0.f;

  for (int b = 0; b < B_; ++b) {
    const float* zb  = z  + (size_t)(b * T_ + t) * N_ * D_;
    const float* sqb = sq + (size_t)(b * T_ + t) * N_;

    // stage squared norms for this (b,t)
    for (int i = tid; i < N_; i += 256) sq_s[i] = sqb[i];
    __syncthreads();

    // ---- dot = z zT : each wave -> 4 col tiles of 16 ----
    v8f c0 = {}, c1 = {}, c2 = {}, c3 = {};
    const float* arp = zb + (size_t)arow * D_;
    const float* bp0 = zb + (size_t)((wave * 4 + 0) * 16 + l15) * D_;
    const float* bp1 = zb + (size_t)((wave * 4 + 1) * 16 + l15) * D_;
    const float* bp2 = zb + (size_t)((wave * 4 + 2) * 16 + l15) * D_;
    const float* bp3 = zb + (size_t)((wave * 4 + 3) * 16 + l15) * D_;
#pragma unroll
    for (int k = 0; k < D_; k += 4) {
      const v2f a = *(const v2f*)(arp + k + koff);
      v2f bm;
      bm = *(const v2f*)(bp0 + k + koff);
      c0 = __builtin_amdgcn_wmma_f32_16x16x4_f32(false, a, false, bm, (short)0, c0, false, false);
      bm = *(const v2f*)(bp1 + k + koff);
      c1 = __builtin_amdgcn_wmma_f32_16x16x4_f32(false, a, false, bm, (short)0, c1, false, false);
      bm = *(const v2f*)(bp2 + k + koff);
      c2 = __builtin_amdgcn_wmma_f32_16x16x4_f32(false, a, false, bm, (short)0, c2, false, false);
      bm = *(const v2f*)(bp3 + k + koff);
      c3 = __builtin_amdgcn_wmma_f32_16x16x4_f32(false, a, false, bm, (short)0, c3, false, false);
    }

    // ---- fused elementwise: att = exp(cos - ||zi-zj|| / (|zi|+|zj|+eps)) ----
#pragma unroll
    for (int ct = 0; ct < 4; ++ct) {
      const v8f c = (ct == 0) ? c0 : (ct == 1) ? c1 : (ct == 2) ? c2 : c3;
      const int col = (wave * 4 + ct) * 16 + l15;
      const float sqm = sq_s[col];
      const float nm  = sqrtf(sqm) + EPS_;
#pragma unroll
      for (int v = 0; v < 8; ++v) {
        const int rloc = rbase + v;
        const float sqn = sq_s[ntile * 16 + rloc];
        const float nn  = sqrtf(sqn) + EPS_;
        const float dot = c[v];
        const float cosv = dot / (nn * nm);
        const float d2   = fmaxf(sqn + sqm - 2.0f * dot, 0.0f);
        const float diff = sqrtf(d2);
        att_s[rloc * 512 + col] = __expf(cosv - diff / (nn + nm + EPS_));
      }
    }
    __syncthreads();

    // ---- row sums: 16 threads per row, 32 strided elements each ----
    {
      const int row = tid >> 4;
      const int sl  = tid & 15;
      float part = 0.f;
#pragma unroll 4
      for (int cidx = sl; cidx < 512; cidx += 16) part += att_s[row * 512 + cidx];
      red_s[tid] = part;
    }
    __syncthreads();
    if ((tid & 15) == 0) {
      const int row = tid >> 4;
      float s = 0.f;
#pragma unroll
      for (int j = 0; j < 16; ++j) s += red_s[(row << 4) + j];
      rowsum_s[row] = s;
    }
    __syncthreads();

    // ---- accumulate mean over B in registers ----
    const float invB = 1.0f / (float)B_;
#pragma unroll
    for (int j = 0; j < 32; ++j) {
      const int idx = tid + j * 256;
      const int r   = idx >> 9;
      acc[j] += att_s[idx] * invB / (rowsum_s[r] + EPS_);
    }
    __syncthreads();   // att_s/rowsum_s reused next b iteration
  }

  // ---- write out[t, ntile*16 + r, m] ----
  float* outp = out + ((size_t)t * N_ + ntile * 16) * N_;
#pragma unroll
  for (int j = 0; j < 32; ++j) {
    const int idx = tid + j * 256;
    outp[idx] = acc[j];
  }
}

// ---------------------------------------------------------------------------
extern "C" void kernel_launch(void* const* d_in, const int* in_sizes, int n_in,
                              void* d_out, int out_size, void* d_ws, size_t ws_size,
                              hipStream_t stream) {
  const float* x    = (const float*)d_in[0];   // [B,N,F,T]
  const float* W    = (const float*)d_in[1];   // [D,F]
  const float* bias = (const float*)d_in[2];   // [D]
  float* out = (float*)d_out;                  // [T,N,N]

  float* z  = (float*)d_ws;                            // B*T*N*D floats (33.5 MB)
  float* sq = z + (size_t)B_ * T_ * N_ * D_;           // B*T*N floats

  // 1) z = tanh(x^T W^T + b): 32768 wave-tiles, 4 waves/block
  GAT_z_kernel<<<8192, 128, 0, stream>>>(x, W, bias, z);

  // 2) squared norms
  GAT_sq_kernel<<<(B_ * T_ * N_ + 255) / 256, 256, 0, stream>>>(z, sq);

  // 3) attention + softmax + mean over B
  GAT_att_kernel<<<T_ * (N_ / 16), 256, 0, stream>>>(z, sq, out);
}